// StableTTLayer_3753801417457
// MI455X (gfx1250) — compile-verified
//
#include <hip/hip_runtime.h>

#define BATCH   131072
#define ORDER   8
#define NDIM    64
#define RDIM    32
#define BLOCK   256                 // threads per block (8 waves)
#define SPT     2                   // samples per thread
#define TILE    (BLOCK * SPT)       // 512 samples per workgroup
#define NWAVES  (BLOCK / 32)
#define VSTRIDE 33                  // 32 + 1 pad -> conflict-free LDS row gathers
#define VROWS   (TILE + 1)          // +1 dummy row for discarded pad writes
#define MSTRIDE (NDIM * RDIM)       // row stride (floats) inside one mid core
#define SMEM_BYTES (2 * VROWS * VSTRIDE * 4)

typedef float v2f __attribute__((ext_vector_type(2)));
typedef float v8f __attribute__((ext_vector_type(8)));

__global__ __launch_bounds__(BLOCK) void tt_forward_kernel(
    const int*   __restrict__ indices,    // [B][ORDER]
    const float* __restrict__ core0,      // [1][N][R]
    const float* __restrict__ cores_mid,  // [ORDER-2][R][N][R]
    const float* __restrict__ core_last,  // [R][N][1]
    float*       __restrict__ out)        // [B]
{
    extern __shared__ float smem[];       // two v buffers (dynamic LDS)
    float* vbuf0 = smem;
    float* vbuf1 = smem + VROWS * VSTRIDE;

    __shared__ unsigned perm[TILE];
    __shared__ unsigned bcnt[NDIM];
    __shared__ unsigned boff[NDIM];

    const int  t    = threadIdx.x;
    const int  lane = t & 31;
    const int  wave = t >> 5;
    const long base = (long)blockIdx.x * TILE;   // BATCH % TILE == 0 -> no guards

    // ---- Phase 0: v = core0[0, idx[s,0], :] for this block's TILE samples ----
    #pragma unroll
    for (int j = 0; j < SPT; ++j) {
        const int  tt = t + j * BLOCK;
        const long s  = base + tt;
        int i0 = indices[s * ORDER + 0];
        i0 = min(max(i0, 0), NDIM - 1);
        const float4* src = (const float4*)(core0 + (long)i0 * RDIM);
        float* dst = &vbuf0[tt * VSTRIDE];
        #pragma unroll
        for (int q = 0; q < RDIM / 4; ++q) {
            float4 x = src[q];
            dst[q * 4 + 0] = x.x; dst[q * 4 + 1] = x.y;
            dst[q * 4 + 2] = x.z; dst[q * 4 + 3] = x.w;
        }
    }
    __syncthreads();

    float* vcur = vbuf0;
    float* vnxt = vbuf1;

    for (int step = 0; step < ORDER - 2; ++step) {
        // ---- histogram of this step's indices over 64 buckets ----
        if (t < NDIM) bcnt[t] = 0;
        __syncthreads();

        int dval[SPT]; unsigned posv[SPT];
        #pragma unroll
        for (int j = 0; j < SPT; ++j) {
            const long s = base + t + j * BLOCK;
            int d = indices[s * ORDER + (step + 1)];
            d = min(max(d, 0), NDIM - 1);
            dval[j] = d;
            posv[j] = atomicAdd(&bcnt[d], 1u);
        }
        __syncthreads();

        // ---- exclusive prefix sum (wave 0, shuffle scan: 2 buckets/lane) ----
        if (wave == 0) {
            const unsigned a0 = bcnt[2 * lane + 0];
            const unsigned a1 = bcnt[2 * lane + 1];
            const unsigned sum = a0 + a1;
            unsigned sc = sum;
            #pragma unroll
            for (int o = 1; o < 32; o <<= 1) {
                const unsigned u = __shfl_up(sc, o, 32);
                if (lane >= o) sc += u;
            }
            const unsigned excl = sc - sum;
            boff[2 * lane + 0] = excl;
            boff[2 * lane + 1] = excl + a0;
        }
        // warm L2 with the next step's core while scanning
        if (step + 1 < ORDER - 2) {
            const float* nxt = cores_mid + (long)(step + 1) * RDIM * MSTRIDE;
            for (int q = t; q < (RDIM * MSTRIDE) / 32; q += BLOCK)
                __builtin_prefetch(nxt + q * 32, 0, 1);
        }
        __syncthreads();

        // ---- scatter permutation: samples of equal index become contiguous ----
        #pragma unroll
        for (int j = 0; j < SPT; ++j)
            perm[boff[dval[j]] + posv[j]] = (unsigned)(t + j * BLOCK);
        __syncthreads();

        const float* Mbase = cores_mid + (long)step * RDIM * MSTRIDE;

        // ---- each wave processes its buckets with f32 WMMA tiles ----
        for (int b = wave; b < NDIM; b += NWAVES) {
            const unsigned c = bcnt[b];
            if (c == 0) continue;
            const unsigned off = boff[b];
            const float* M = Mbase + (long)b * RDIM;  // M[r][col] at M[r*MSTRIDE + col]

            // B fragments (shared by all tiles of this bucket): 8 K-tiles x 2 N-tiles
            v2f bf[8][2];
            {
                const int scol0 = lane & 15;
                const int rbase = (lane < 16) ? 0 : 2;
                #pragma unroll
                for (int kk = 0; kk < 8; ++kk) {
                    const int r = kk * 4 + rbase;
                    #pragma unroll
                    for (int nn = 0; nn < 2; ++nn) {
                        const int scol = nn * 16 + scol0;
                        bf[kk][nn][0] = M[(long)r       * MSTRIDE + scol];
                        bf[kk][nn][1] = M[(long)(r + 1) * MSTRIDE + scol];
                    }
                }
            }

            const int col   = lane & 15;
            const int nhalf = (lane < 16) ? 0 : 8;
            const int koffb = (lane < 16) ? 0 : 2;

            for (unsigned t0 = 0; t0 < c; t0 += 16) {
                // A rows: pad rows read a valid row; their results go to the dummy row
                const unsigned aidx = t0 + (unsigned)col;
                const unsigned arow = perm[off + ((aidx < c) ? aidx : 0u)];
                const float* abase = &vcur[arow * VSTRIDE + koffb];

                // 4 independent WMMA chains (even/odd K split) to cover WMMA latency
                v8f acc00 = {}; v8f acc01 = {};   // N-tile 0: even-K chain, odd-K chain
                v8f acc10 = {}; v8f acc11 = {};   // N-tile 1: even-K chain, odd-K chain
                #pragma unroll
                for (int kk = 0; kk < 8; kk += 2) {
                    v2f a0, a1;
                    const float* ap0 = abase + kk * 4;
                    a0[0] = ap0[0]; a0[1] = ap0[1];
                    a1[0] = ap0[4]; a1[1] = ap0[5];
                    acc00 = __builtin_amdgcn_wmma_f32_16x16x4_f32(
                        false, a0, false, bf[kk][0],     (short)0, acc00, false, false);
                    acc10 = __builtin_amdgcn_wmma_f32_16x16x4_f32(
                        false, a0, false, bf[kk][1],     (short)0, acc10, false, false);
                    acc01 = __builtin_amdgcn_wmma_f32_16x16x4_f32(
                        false, a1, false, bf[kk + 1][0], (short)0, acc01, false, false);
                    acc11 = __builtin_amdgcn_wmma_f32_16x16x4_f32(
                        false, a1, false, bf[kk + 1][1], (short)0, acc11, false, false);
                }
                const v8f acc0 = acc00 + acc01;
                const v8f acc1 = acc10 + acc11;

                // branchless scatter of D rows (invalid rows -> dummy row TILE)
                #pragma unroll
                for (int i = 0; i < 8; ++i) {
                    const unsigned idxb = t0 + (unsigned)(i + nhalf);
                    unsigned slot = perm[off + ((idxb < c) ? idxb : 0u)];
                    slot = (idxb < c) ? slot : (unsigned)TILE;
                    float* p = &vnxt[slot * VSTRIDE + col];
                    p[0]  = acc0[i];
                    p[16] = acc1[i];
                }
            }
        }
        __syncthreads();
        float* tmp = vcur; vcur = vnxt; vnxt = tmp;
    }

    // ---- final: out[s] = dot(v, core_last[:, idx7, 0]) ----
    #pragma unroll
    for (int j = 0; j < SPT; ++j) {
        const int  tt = t + j * BLOCK;
        const long s  = base + tt;
        int d7 = indices[s * ORDER + (ORDER - 1)];
        d7 = min(max(d7, 0), NDIM - 1);
        const float* v = &vcur[tt * VSTRIDE];
        float acc = 0.0f;
        #pragma unroll
        for (int r = 0; r < RDIM; ++r)
            acc = fmaf(v[r], core_last[(long)r * NDIM + d7], acc);
        out[s] = acc;
    }
}

extern "C" void kernel_launch(void* const* d_in, const int* in_sizes, int n_in,
                              void* d_out, int out_size, void* d_ws, size_t ws_size,
                              hipStream_t stream) {
    const int*   indices   = (const int*)  d_in[0];
    const float* core0     = (const float*)d_in[1];
    const float* cores_mid = (const float*)d_in[2];
    const float* core_last = (const float*)d_in[3];
    float* outp = (float*)d_out;

    (void)hipFuncSetAttribute((const void*)tt_forward_kernel,
                              hipFuncAttributeMaxDynamicSharedMemorySize,
                              SMEM_BYTES);

    const int groups = BATCH / TILE;   // 256, exact
    hipLaunchKernelGGL(tt_forward_kernel, dim3(groups), dim3(BLOCK), SMEM_BYTES, stream,
                       indices, core0, cores_mid, core_last, outp);
    (void)in_sizes; (void)n_in; (void)out_size; (void)d_ws; (void)ws_size;
}